// PropagationModel_90864328114273
// MI455X (gfx1250) — compile-verified
//
#include <hip/hip_runtime.h>
#include <stdint.h>
#include <math.h>

#define NNODES 550000
#define NEDGES 5000000
#define EPSB   1e-5f

typedef __attribute__((ext_vector_type(2))) float v2f;
typedef __attribute__((ext_vector_type(8))) float v8f;

// ---- monotonic float<->uint mapping for atomicMax-based segment max ----
__device__ __forceinline__ unsigned fflip(float f) {
  unsigned u = __float_as_uint(f);
  return (u & 0x80000000u) ? ~u : (u | 0x80000000u);
}
__device__ __forceinline__ float funflip(unsigned u) {
  return (u & 0x80000000u) ? __uint_as_float(u ^ 0x80000000u)
                           : __uint_as_float(~u);
}

// ======================= Layer-1 transform (WMMA, K=2 padded to 4) ==========
// xl = x@Wl + bl ; xr = x@Wr + br   for a 16-node tile per wave.
__global__ void __launch_bounds__(256)
k_transform1(const float* __restrict__ x,
             const float* __restrict__ Wl, const float* __restrict__ bl,
             const float* __restrict__ Wr, const float* __restrict__ br,
             float* __restrict__ xl, float* __restrict__ xr, int ntiles)
{
  int wave = blockIdx.x * (blockDim.x >> 5) + (threadIdx.x >> 5);
  if (wave >= ntiles) return;                 // whole-wave exit, EXEC stays full
  int lane  = threadIdx.x & 31;
  int col   = lane & 15;                      // B column / A row (M)
  int khalf = lane >> 4;                      // K half select
  int grow  = wave * 16 + col;

  // A fragment (16x4): K=0,1 live in lanes 0-15; K=2,3 (zero pad) in 16-31
  v2f a;
  a.x = khalf ? 0.f : x[(size_t)grow * 2 + 0];
  a.y = khalf ? 0.f : x[(size_t)grow * 2 + 1];
  // B fragments (4x16): rows 0,1 = W, rows 2,3 = 0
  v2f bL, bR;
  bL.x = khalf ? 0.f : Wl[col];      bL.y = khalf ? 0.f : Wl[16 + col];
  bR.x = khalf ? 0.f : Wr[col];      bR.y = khalf ? 0.f : Wr[16 + col];

  v8f cl, cr;
  float biasl = bl[col], biasr = br[col];
#pragma unroll
  for (int j = 0; j < 8; j++) { cl[j] = biasl; cr[j] = biasr; }

  cl = __builtin_amdgcn_wmma_f32_16x16x4_f32(false, a, false, bL, (short)0, cl, false, false);
  cr = __builtin_amdgcn_wmma_f32_16x16x4_f32(false, a, false, bR, (short)0, cr, false, false);

  int base = wave * 16;
#pragma unroll
  for (int j = 0; j < 8; j++) {
    int r = base + j + 8 * khalf;             // C/D layout: VGPR j -> rows j, j+8
    xl[(size_t)r * 16 + col] = cl[j];
    xr[(size_t)r * 16 + col] = cr[j];
  }
}

// ======================= Edge pass 1: per-destination segment max ===========
__global__ void __launch_bounds__(256)
k_edge_max(const int* __restrict__ ei,
           const float* __restrict__ xl, const float* __restrict__ xr,
           const float* __restrict__ att,
           unsigned* __restrict__ maxb)
{
  long long idx = (long long)blockIdx.x * blockDim.x + threadIdx.x;
  long long tot = (long long)NEDGES + NNODES;
  if (idx >= tot) return;
  int s, d;
  if (idx < NEDGES) {                          // streaming reads: don't pollute L2
    s = __builtin_nontemporal_load(&ei[idx]);
    d = __builtin_nontemporal_load(&ei[NEDGES + idx]);
  } else {
    s = d = (int)(idx - NEDGES);               // appended self-loops
  }

  float ls[16], rd[16];
  const float4* ps = (const float4*)(xl + (size_t)s * 16);
  const float4* pd = (const float4*)(xr + (size_t)d * 16);
#pragma unroll
  for (int i = 0; i < 4; i++) {
    float4 t = ps[i]; ls[4*i]=t.x; ls[4*i+1]=t.y; ls[4*i+2]=t.z; ls[4*i+3]=t.w;
    float4 u = pd[i]; rd[4*i]=u.x; rd[4*i+1]=u.y; rd[4*i+2]=u.z; rd[4*i+3]=u.w;
  }
#pragma unroll
  for (int h = 0; h < 8; h++) {
    float v0 = ls[2*h]   + rd[2*h];
    float v1 = ls[2*h+1] + rd[2*h+1];
    v0 = v0 > 0.f ? v0 : 0.2f * v0;           // leaky_relu(.,0.2)
    v1 = v1 > 0.f ? v1 : 0.2f * v1;
    float lg = v0 * att[2*h] + v1 * att[2*h+1];
    atomicMax(&maxb[(size_t)d * 8 + h], fflip(lg));
  }
}

// ======= Edge pass 2: exp(logit-max), accumulate denom and numerator ========
__global__ void __launch_bounds__(256)
k_edge_sum(const int* __restrict__ ei,
           const float* __restrict__ xl, const float* __restrict__ xr,
           const float* __restrict__ att,
           const unsigned* __restrict__ maxb,
           float* __restrict__ denom, float* __restrict__ num)
{
  long long idx = (long long)blockIdx.x * blockDim.x + threadIdx.x;
  long long tot = (long long)NEDGES + NNODES;
  if (idx >= tot) return;
  int s, d;
  if (idx < NEDGES) {
    s = __builtin_nontemporal_load(&ei[idx]);
    d = __builtin_nontemporal_load(&ei[NEDGES + idx]);
  } else {
    s = d = (int)(idx - NEDGES);
  }

  float ls[16], rd[16];
  const float4* ps = (const float4*)(xl + (size_t)s * 16);
  const float4* pd = (const float4*)(xr + (size_t)d * 16);
#pragma unroll
  for (int i = 0; i < 4; i++) {
    float4 t = ps[i]; ls[4*i]=t.x; ls[4*i+1]=t.y; ls[4*i+2]=t.z; ls[4*i+3]=t.w;
    float4 u = pd[i]; rd[4*i]=u.x; rd[4*i+1]=u.y; rd[4*i+2]=u.z; rd[4*i+3]=u.w;
  }
#pragma unroll
  for (int h = 0; h < 8; h++) {
    float v0 = ls[2*h]   + rd[2*h];
    float v1 = ls[2*h+1] + rd[2*h+1];
    v0 = v0 > 0.f ? v0 : 0.2f * v0;
    v1 = v1 > 0.f ? v1 : 0.2f * v1;
    float lg = v0 * att[2*h] + v1 * att[2*h+1];
    float mx = funflip(maxb[(size_t)d * 8 + h]);
    float e  = expf(lg - mx);
    atomicAdd(&denom[(size_t)d * 8 + h], e);
    atomicAdd(&num[(size_t)d * 16 + 2*h    ], e * ls[2*h]);
    atomicAdd(&num[(size_t)d * 16 + 2*h + 1], e * ls[2*h+1]);
  }
}

// ===== Finish a GAT layer: out = num/denom + gbias; accumulate BN stats =====
__global__ void __launch_bounds__(256)
k_finish(const float* __restrict__ num, const float* __restrict__ denom,
         const float* __restrict__ gbias, float* __restrict__ hout,
         double* __restrict__ gsum, double* __restrict__ gsq)
{
  __shared__ double lsum[16], lsq[16];
  int t = threadIdx.x;
  if (t < 16) { lsum[t] = 0.0; lsq[t] = 0.0; }
  __syncthreads();
  int n = blockIdx.x * blockDim.x + t;
  if (n < NNODES) {
#pragma unroll
    for (int f = 0; f < 16; f++) {
      float v = num[(size_t)n*16 + f] / denom[(size_t)n*8 + (f >> 1)] + gbias[f];
      hout[(size_t)n*16 + f] = v;
      atomicAdd(&lsum[f], (double)v);
      atomicAdd(&lsq[f],  (double)v * (double)v);
    }
  }
  __syncthreads();
  if (t < 16) { atomicAdd(&gsum[t], lsum[t]); atomicAdd(&gsq[t], lsq[t]); }
}

// ======= Layer-2 transform: fused BN1 + [16x16]x[16x16] via 4 chained WMMA ==
__global__ void __launch_bounds__(256)
k_transform2(const float* __restrict__ h1,
             const double* __restrict__ gsum, const double* __restrict__ gsq,
             const float* __restrict__ bng, const float* __restrict__ bnb,
             const float* __restrict__ Wl, const float* __restrict__ bl,
             const float* __restrict__ Wr, const float* __restrict__ br,
             float* __restrict__ xl, float* __restrict__ xr, int ntiles)
{
  int wave = blockIdx.x * (blockDim.x >> 5) + (threadIdx.x >> 5);
  if (wave >= ntiles) return;
  int lane  = threadIdx.x & 31;
  int col   = lane & 15;
  int khalf = lane >> 4;
  int grow  = wave * 16 + col;
  const float invN = 1.0f / (float)NNODES;

  v2f af[4], bfl[4], bfr[4];
#pragma unroll
  for (int kc = 0; kc < 4; kc++) {
    int k0 = kc * 4 + 2 * khalf, k1 = k0 + 1;
    float mu0 = (float)gsum[k0] * invN, mu1 = (float)gsum[k1] * invN;
    float rs0 = rsqrtf((float)gsq[k0] * invN - mu0*mu0 + EPSB);
    float rs1 = rsqrtf((float)gsq[k1] * invN - mu1*mu1 + EPSB);
    af[kc].x = (h1[(size_t)grow*16 + k0] - mu0) * rs0 * bng[k0] + bnb[k0];
    af[kc].y = (h1[(size_t)grow*16 + k1] - mu1) * rs1 * bng[k1] + bnb[k1];
    bfl[kc].x = Wl[k0*16 + col];  bfl[kc].y = Wl[k1*16 + col];
    bfr[kc].x = Wr[k0*16 + col];  bfr[kc].y = Wr[k1*16 + col];
  }
  v8f cl, cr;
  float b1 = bl[col], b2 = br[col];
#pragma unroll
  for (int j = 0; j < 8; j++) { cl[j] = b1; cr[j] = b2; }
#pragma unroll
  for (int kc = 0; kc < 4; kc++)
    cl = __builtin_amdgcn_wmma_f32_16x16x4_f32(false, af[kc], false, bfl[kc], (short)0, cl, false, false);
#pragma unroll
  for (int kc = 0; kc < 4; kc++)
    cr = __builtin_amdgcn_wmma_f32_16x16x4_f32(false, af[kc], false, bfr[kc], (short)0, cr, false, false);

  int base = wave * 16;
#pragma unroll
  for (int j = 0; j < 8; j++) {
    int r = base + j + 8 * khalf;
    xl[(size_t)r*16 + col] = cl[j];
    xr[(size_t)r*16 + col] = cr[j];
  }
}

// ===== BN2 + ReLU + 16->2 linear; accumulate BN3 stats ======================
__global__ void __launch_bounds__(256)
k_final_pre(const float* __restrict__ h2,
            const double* __restrict__ gsum2, const double* __restrict__ gsq2,
            const float* __restrict__ g2, const float* __restrict__ b2,
            const float* __restrict__ linW, const float* __restrict__ linb,
            float* __restrict__ y, double* __restrict__ stats3)
{
  __shared__ double acc[4];
  int t = threadIdx.x;
  if (t < 4) acc[t] = 0.0;
  __syncthreads();
  int n = blockIdx.x * blockDim.x + t;
  if (n < NNODES) {
    const float invN = 1.0f / (float)NNODES;
    float y0 = linb[0], y1 = linb[1];
#pragma unroll
    for (int f = 0; f < 16; f++) {
      float mu = (float)gsum2[f] * invN;
      float rs = rsqrtf((float)gsq2[f] * invN - mu*mu + EPSB);
      float v  = (h2[(size_t)n*16 + f] - mu) * rs * g2[f] + b2[f];
      v = v > 0.f ? v : 0.f;                     // dropout==identity, relu
      y0 += v * linW[f*2 + 0];
      y1 += v * linW[f*2 + 1];
    }
    y[(size_t)n*2 + 0] = y0;
    y[(size_t)n*2 + 1] = y1;
    atomicAdd(&acc[0], (double)y0); atomicAdd(&acc[1], (double)y0 * (double)y0);
    atomicAdd(&acc[2], (double)y1); atomicAdd(&acc[3], (double)y1 * (double)y1);
  }
  __syncthreads();
  if (t < 4) atomicAdd(&stats3[t], acc[t]);
}

// ===== BN3 + 11-node group mean =============================================
__global__ void __launch_bounds__(256)
k_final_out(const float* __restrict__ y, const double* __restrict__ stats3,
            const float* __restrict__ g3, const float* __restrict__ b3,
            float* __restrict__ out)
{
  int g = blockIdx.x * blockDim.x + threadIdx.x;
  if (g >= NNODES / 11) return;
  const float invN = 1.0f / (float)NNODES;
  float mu0 = (float)stats3[0] * invN, mu1 = (float)stats3[2] * invN;
  float rs0 = rsqrtf((float)stats3[1] * invN - mu0*mu0 + EPSB);
  float rs1 = rsqrtf((float)stats3[3] * invN - mu1*mu1 + EPSB);
  float a0 = 0.f, a1 = 0.f;
  for (int i = 0; i < 11; i++) {
    int n = g * 11 + i;
    float v0 = __builtin_nontemporal_load(&y[(size_t)n*2 + 0]);
    float v1 = __builtin_nontemporal_load(&y[(size_t)n*2 + 1]);
    a0 += (v0 - mu0) * rs0 * g3[0] + b3[0];
    a1 += (v1 - mu1) * rs1 * g3[1] + b3[1];
  }
  out[g*2 + 0] = a0 * (1.0f / 11.0f);
  out[g*2 + 1] = a1 * (1.0f / 11.0f);
}

// ============================================================================
extern "C" void kernel_launch(void* const* d_in, const int* in_sizes, int n_in,
                              void* d_out, int out_size, void* d_ws, size_t ws_size,
                              hipStream_t stream)
{
  const float* x    = (const float*)d_in[0];
  const float* Wl1  = (const float*)d_in[1];
  const float* bl1  = (const float*)d_in[2];
  const float* Wr1  = (const float*)d_in[3];
  const float* br1  = (const float*)d_in[4];
  const float* att1 = (const float*)d_in[5];
  const float* gb1  = (const float*)d_in[6];
  const float* bn1g = (const float*)d_in[7];
  const float* bn1b = (const float*)d_in[8];
  const float* Wl2  = (const float*)d_in[9];
  const float* bl2  = (const float*)d_in[10];
  const float* Wr2  = (const float*)d_in[11];
  const float* br2  = (const float*)d_in[12];
  const float* att2 = (const float*)d_in[13];
  const float* gb2  = (const float*)d_in[14];
  const float* bn2g = (const float*)d_in[15];
  const float* bn2b = (const float*)d_in[16];
  const float* linW = (const float*)d_in[17];
  const float* linb = (const float*)d_in[18];
  const float* bn3g = (const float*)d_in[19];
  const float* bn3b = (const float*)d_in[20];
  const int*   ei   = (const int*)d_in[21];

  const size_t N = NNODES;
  float*    xl    = (float*)d_ws;
  float*    xr    = xl  + 16 * N;
  float*    num   = xr  + 16 * N;
  float*    h     = num + 16 * N;     // h1, later reused for h2
  float*    denom = h   + 16 * N;
  unsigned* maxb  = (unsigned*)(denom + 8 * N);
  float*    y     = (float*)(maxb + 8 * N);
  double*   stats = (double*)(((uintptr_t)(y + 2 * N) + 15) & ~(uintptr_t)15);
  double* s1sum = stats;       double* s1sq = stats + 16;
  double* s2sum = stats + 32;  double* s2sq = stats + 48;
  double* s3    = stats + 64;  // 4 doubles

  // zero accumulators (graph-capture-safe memset nodes)
  hipMemsetAsync(num,   0, 16 * N * sizeof(float),    stream);
  hipMemsetAsync(denom, 0,  8 * N * sizeof(float),    stream);
  hipMemsetAsync(maxb,  0,  8 * N * sizeof(unsigned), stream);
  hipMemsetAsync(stats, 0, 68 * sizeof(double),       stream);

  dim3 tb(256);
  const int ntiles  = NNODES / 16;                 // 34375 exact
  const int tblocks = (ntiles + 7) / 8;            // 8 waves/block
  const long long tot = (long long)NEDGES + NNODES;
  const int eblocks = (int)((tot + 255) / 256);
  const int nblocks = (NNODES + 255) / 256;

  // -------- layer 1 --------
  k_transform1<<<tblocks, tb, 0, stream>>>(x, Wl1, bl1, Wr1, br1, xl, xr, ntiles);
  k_edge_max  <<<eblocks, tb, 0, stream>>>(ei, xl, xr, att1, maxb);
  k_edge_sum  <<<eblocks, tb, 0, stream>>>(ei, xl, xr, att1, maxb, denom, num);
  k_finish    <<<nblocks, tb, 0, stream>>>(num, denom, gb1, h, s1sum, s1sq);

  // -------- layer 2 (BN1 fused into transform) --------
  k_transform2<<<tblocks, tb, 0, stream>>>(h, s1sum, s1sq, bn1g, bn1b,
                                           Wl2, bl2, Wr2, br2, xl, xr, ntiles);
  hipMemsetAsync(num,   0, 16 * N * sizeof(float),    stream);
  hipMemsetAsync(denom, 0,  8 * N * sizeof(float),    stream);
  hipMemsetAsync(maxb,  0,  8 * N * sizeof(unsigned), stream);
  k_edge_max  <<<eblocks, tb, 0, stream>>>(ei, xl, xr, att2, maxb);
  k_edge_sum  <<<eblocks, tb, 0, stream>>>(ei, xl, xr, att2, maxb, denom, num);
  k_finish    <<<nblocks, tb, 0, stream>>>(num, denom, gb2, h, s2sum, s2sq);

  // -------- BN2 + relu + linear + BN3 + group mean --------
  k_final_pre <<<nblocks, tb, 0, stream>>>(h, s2sum, s2sq, bn2g, bn2b,
                                           linW, linb, y, s3);
  const int gblocks = (NNODES / 11 + 255) / 256;
  k_final_out <<<gblocks, tb, 0, stream>>>(y, s3, bn3g, bn3b, (float*)d_out);
}